// GnnLastLayer_18691697672392
// MI455X (gfx1250) — compile-verified
//
#include <hip/hip_runtime.h>

typedef __attribute__((ext_vector_type(2))) float v2f;
typedef __attribute__((ext_vector_type(4))) float v4f;
typedef __attribute__((ext_vector_type(8))) float v8f;

#define BM 64
#define BK 32
#define PA 36    // A-tile LDS pitch (floats): conflict-free 16-lane b64 frag reads
#define PBP 288  // interleaved-B pair-row pitch: 288 % 64 == 32 -> half-wave rows on disjoint banks

// ---------------------------------------------------------------------------
// fp32 WMMA GEMM, C_tile[64 x 128] = A[64 x klen] * B[klen x 128]
//   A : row-major, leading dim lda
//   Bp: k-pair interleaved, rows = Kd/2, each row 256 floats:
//       Bp[kp][col*2 + j] == B[2*kp + j][col]   (fragment = one ds_load_b64)
//   C : element (m, col) stored at Cy + m*c_row_stride + col*c_col_stride,
//       Cy = Cb + (y>>1)*c_y_hi + (y&1)*c_y_lo   (y = blockIdx.y)
//   k-range for this block: [y*k_y_step, y*k_y_step + klen)
// Ping-pong double-buffered LDS, one barrier per BK chunk.
// ---------------------------------------------------------------------------
__global__ __launch_bounds__(256) void wmma_gemm_f32(
    const float* __restrict__ A, int lda,
    const float* __restrict__ Bp, int b_y_stride,
    float* __restrict__ Cb, int c_y_hi, int c_y_lo,
    int c_row_stride, int c_col_stride,
    int k_y_step, int klen)
{
  __shared__ float Alds[2][BM * PA];
  __shared__ float Blds[2][16 * PBP];

  const int y = blockIdx.y;
  const float* B = Bp + (size_t)y * b_y_stride;
  float* C = Cb + (size_t)(y >> 1) * c_y_hi + (size_t)(y & 1) * c_y_lo;
  const int kbase0 = y * k_y_step;
  const int kend = kbase0 + klen;

  const int t    = threadIdx.x;
  const int lane = t & 31;
  const int wid  = t >> 5;        // 0..7
  const int wm   = wid >> 1;      // 0..3 : 16-row strip
  const int wn   = wid & 1;       // 0..1 : 64-col half
  const int lm   = lane & 15;
  const int hi   = (lane >= 16) ? 1 : 0;   // lanes 16-31 carry K=2,3 (ISA fp32 layout)
  const int rowbase = blockIdx.x * BM;

  // cooperative-load thread mapping (fully coalesced b128)
  const int ar  = t >> 2;            // 0..63  A row
  const int aq  = (t & 3) * 8;       // 0,8,16,24
  const int bkp = t >> 4;            // 0..15  B pair-row
  const int bsg = (t & 15) * 16;     // 0..240

  const float* ga = A + (size_t)(rowbase + ar) * lda + aq;   // + kb
  const float* gb = B + (size_t)bkp * 256 + bsg;             // + kb*128 (== (kb/2)*256)

  float* sA[2] = { &Alds[0][ar * PA + aq],  &Alds[1][ar * PA + aq]  };
  float* sB[2] = { &Blds[0][bkp * PBP + bsg], &Blds[1][bkp * PBP + bsg] };

  v8f acc[4];
  #pragma unroll
  for (int i = 0; i < 4; ++i) acc[i] = (v8f){0.f,0.f,0.f,0.f,0.f,0.f,0.f,0.f};

  // ---- preload chunk 0 into LDS buffer 0 ----
  v4f ra0 = *(const v4f*)(ga + kbase0);
  v4f ra1 = *(const v4f*)(ga + kbase0 + 4);
  const float* gb0 = gb + (size_t)kbase0 * 128;
  v4f rb0 = *(const v4f*)(gb0);
  v4f rb1 = *(const v4f*)(gb0 + 4);
  v4f rb2 = *(const v4f*)(gb0 + 8);
  v4f rb3 = *(const v4f*)(gb0 + 12);
  *(v4f*)(sA[0])     = ra0;  *(v4f*)(sA[0] + 4) = ra1;
  *(v4f*)(sB[0])     = rb0;  *(v4f*)(sB[0] + 4) = rb1;
  *(v4f*)(sB[0] + 8) = rb2;  *(v4f*)(sB[0] + 12) = rb3;
  __syncthreads();

  int buf = 0;
  for (int kb = kbase0; kb < kend; kb += BK) {
    const int  kn   = kb + BK;
    const bool more = kn < kend;   // uniform across block
    if (more) {
      // register-prefetch next chunk; latency hidden behind the 32 WMMAs below
      ra0 = *(const v4f*)(ga + kn);
      ra1 = *(const v4f*)(ga + kn + 4);
      const float* gbn = gb + (size_t)kn * 128;
      rb0 = *(const v4f*)(gbn);
      rb1 = *(const v4f*)(gbn + 4);
      rb2 = *(const v4f*)(gbn + 8);
      rb3 = *(const v4f*)(gbn + 12);
      // keep the HBM A-stream deep: L2 prefetch two chunks ahead
      __builtin_prefetch(ga + kn + 2 * BK, 0, 3);
    }

    const float* Acur = &Alds[buf][0];
    const float* Bcur = &Blds[buf][0];
    #pragma unroll
    for (int ks = 0; ks < BK; ks += 4) {
      // A fragment: lane<16 -> {A[m][ks],A[m][ks+1]}, lane>=16 -> K=2,3
      const float* ap = &Acur[(wm * 16 + lm) * PA + ks + 2 * hi];
      v2f afrag = { ap[0], ap[1] };
      const int kp = (ks >> 1) + hi;   // pair-row: {B[ks+2hi][col], B[ks+2hi+1][col]}
      #pragma unroll
      for (int nt = 0; nt < 4; ++nt) {
        const int col = wn * 64 + nt * 16 + lm;
        v2f bfrag = *(const v2f*)&Bcur[kp * PBP + 2 * col];  // single ds_load_b64
        acc[nt] = __builtin_amdgcn_wmma_f32_16x16x4_f32(
            false, afrag, false, bfrag, (short)0, acc[nt],
            /*reuse_a=*/false, /*reuse_b=*/false);
      }
    }

    if (more) {
      float* dA = sA[buf ^ 1];
      float* dB = sB[buf ^ 1];
      *(v4f*)(dA)     = ra0;  *(v4f*)(dA + 4)  = ra1;
      *(v4f*)(dB)     = rb0;  *(v4f*)(dB + 4)  = rb1;
      *(v4f*)(dB + 8) = rb2;  *(v4f*)(dB + 12) = rb3;
    }
    __syncthreads();
    buf ^= 1;
  }

  // ---- store: C/D layout: VGPR j -> (M = j + 8*hi, N = lm) ----
  #pragma unroll
  for (int nt = 0; nt < 4; ++nt) {
    const int col = wn * 64 + nt * 16 + lm;
    #pragma unroll
    for (int j = 0; j < 8; ++j) {
      const int mrow = rowbase + wm * 16 + j + 8 * hi;
      C[(size_t)mrow * c_row_stride + (size_t)col * c_col_stride] = acc[nt][j];
    }
  }
}

// ---------------------------------------------------------------------------
// One-off gather of fc1_w [128 x 1024] into 4 per-k, f-pair-interleaved blocks:
//   Wtp[k][f>>1][o*2 + (f&1)] = fc1_w[o][f*4+k].  Reads fully coalesced.
// ---------------------------------------------------------------------------
__global__ __launch_bounds__(256) void transpose_w(
    const float* __restrict__ w, float* __restrict__ wt)
{
  const int idx = blockIdx.x * 256 + threadIdx.x;  // 0 .. 131071
  const int o = idx >> 10;
  const int c = idx & 1023;                        // f*4 + k
  const int f = c >> 2;
  const int k = c & 3;
  wt[(size_t)k * 32768 + (size_t)(f >> 1) * 256 + (o << 1) + (f & 1)] = w[idx];
}

// ---------------------------------------------------------------------------
// out[n][o] = fc1_b[o] + sum_{s<8} part[s][n][o]
// ---------------------------------------------------------------------------
__global__ __launch_bounds__(256) void reduce_bias(
    const float* __restrict__ part, const float* __restrict__ bias,
    float* __restrict__ out)
{
  const int idx = blockIdx.x * 256 + threadIdx.x;  // 0 .. 524287
  float s = bias[idx & 127];
  #pragma unroll
  for (int sp = 0; sp < 8; ++sp) s += part[(size_t)sp * 524288 + idx];
  out[idx] = s;
}

extern "C" void kernel_launch(void* const* d_in, const int* in_sizes, int n_in,
                              void* d_out, int out_size, void* d_ws, size_t ws_size,
                              hipStream_t stream) {
  (void)in_sizes; (void)n_in; (void)out_size; (void)ws_size;
  const float* WW    = (const float*)d_in[0];  // [4096,4096,4] == A[4096][16384] row-major
  const float* x     = (const float*)d_in[1];  // [4096,256]
  const float* fc1_w = (const float*)d_in[2];  // [128,1024]
  const float* fc1_b = (const float*)d_in[3];  // [128]
  float* out = (float*)d_out;                  // [4096,128]

  char* ws = (char*)d_ws;
  float* Wtp  = (float*)ws;                                  // 512 KB: 4 x [128][256] pair-interleaved
  float* Yp   = (float*)(ws + (512 << 10));                  // 8 MB:  [8192][256] pair-interleaved
  float* part = (float*)(ws + (512 << 10) + (8 << 20));      // 16 MB: 8 x [4096][128] plain

  // Kernel 0: gather fc1_w into per-k interleaved weight blocks.
  transpose_w<<<512, 256, 0, stream>>>(fc1_w, Wtp);

  // Stage 1: Y[(m*4+k)][o] = sum_f x[m][f] * Wt[k][f][o], stored pair-interleaved:
  //   Yp[(m*4+k)>>1][o*2 + ((m*4+k)&1)]
  //   element (m,o,k=y): addr = (m*2 + (y>>1))*256 + o*2 + (y&1)
  {
    dim3 grid(64, 4);
    wmma_gemm_f32<<<grid, 256, 0, stream>>>(
        x, /*lda=*/256,
        Wtp, /*b_y_stride=*/32768,
        Yp, /*c_y_hi=*/256, /*c_y_lo=*/1,
        /*c_row_stride=*/512, /*c_col_stride=*/2,
        /*k_y_step=*/0, /*klen=*/256);
  }

  // Stage 2: part[s] = WW[:, s*2048:(s+1)*2048] @ Y[s*2048:(s+1)*2048, :]
  //   WW streamed from HBM exactly once; Yp (8 MB) is L2-resident.
  {
    dim3 grid(64, 8);
    wmma_gemm_f32<<<grid, 256, 0, stream>>>(
        WW, /*lda=*/16384,
        Yp, /*b_y_stride=*/0,
        part, /*c_y_hi=*/1048576, /*c_y_lo=*/524288,
        /*c_row_stride=*/128, /*c_col_stride=*/1,
        /*k_y_step=*/2048, /*klen=*/2048);
  }

  // Stage 3: deterministic split-K reduction + bias.
  reduce_bias<<<2048, 256, 0, stream>>>(part, fc1_b, out);
}